// LogicMP_20495583937034
// MI455X (gfx1250) — compile-verified
//
#include <hip/hip_runtime.h>
#include <hip/hip_bf16.h>

// CDNA5 / gfx1250. wave32. Uses V_WMMA_F32_16X16X4_F32 for the dual tile
// reductions (row-sums and a-weighted column-sums of the sigmoid tile).

typedef __attribute__((ext_vector_type(2))) float v2f;
typedef __attribute__((ext_vector_type(8))) float v8f;

// sigmoid with hardware V_RCP_F32 (avoids the precise div_fixup sequence)
__device__ __forceinline__ float sigmoidf(float z) {
    return __builtin_amdgcn_rcpf(1.0f + __expf(-z));
}

// ---------------------------------------------------------------------------
// init: a0[i] = sigmoid(bs1-bs0); zero u, v and the zeros array
// ---------------------------------------------------------------------------
__global__ void lmp_init(const float* __restrict__ bs,
                         float* __restrict__ a0,
                         float* __restrict__ u, float* __restrict__ v,
                         float* __restrict__ zer, int N) {
    int i = blockIdx.x * blockDim.x + threadIdx.x;
    if (i < N) {
        a0[i]  = sigmoidf(bs[2 * i + 1] - bs[2 * i]);
        u[i]   = 0.0f;
        v[i]   = 0.0f;
        zer[i] = 0.0f;
    }
}

// ---------------------------------------------------------------------------
// matvec: for a 16-row stripe (one workgroup, 8 waves splitting column tiles):
//   G[i,j] = sigmoid((bf1-bf0) - w0*ash[i]*(1-ash[j]))
//   u[i] += sum_j G[i,j]*(1-ap[j])      (WMMA: A=G*b tile, B=ones)
//   v[j] += sum_i ap[i]*G[i,j]          (WMMA: A=ones, B=a*G tile)
// base_f (134MB) is L2-resident on MI455X (192MB L2), so the 5 sweeps are
// L2-bandwidth bound, not HBM bound. Phase 2 reads only LDS/registers so the
// WMMA stream never waits on vmem.
// ---------------------------------------------------------------------------
__global__ __launch_bounds__(256) void lmp_matvec(
        const float* __restrict__ bf,     // (N,N,2) interleaved
        const float* __restrict__ ap,     // a_{t-1}
        const float* __restrict__ ash,    // a_{t-2} (zeros at t=1)
        const float* __restrict__ wp,
        float* __restrict__ u, float* __restrict__ v, int N) {
    const int i0   = blockIdx.x << 4;          // 16-row stripe
    const int wave = threadIdx.x >> 5;
    const int lane = threadIdx.x & 31;
    const float w0 = wp[0];

    __shared__ float tile[8][16][17];          // +1 pad: conflict-free strided reads
    __shared__ float wb[8][16];                // per-tile column weights 1-ap[j]

    // phase-1 mapping: lane -> (row r, column half h)
    const int r = lane >> 1;
    const int h = lane & 1;
    const float sh_r = w0 * ash[i0 + r];       // row shift coefficient

    // phase-2 mapping per ISA VGPR layouts (16x4 A / 4x16 B, wave32)
    const int m16  = lane & 15;                // A row / B column
    const int koff = (lane >> 4) << 1;         // lanes>=16 handle K+2,K+3

    // loop-invariant row weights a[i0+4c+koff(+1)] — preload once
    float arow[8];
#pragma unroll
    for (int c = 0; c < 4; ++c) {
        arow[2 * c]     = ap[i0 + 4 * c + koff];
        arow[2 * c + 1] = ap[i0 + 4 * c + koff + 1];
    }

    const v2f ones = {1.0f, 1.0f};
    v8f cu = {};                               // row-sum accumulator (persists)

    const int ntiles = N >> 4;
    for (int ct = wave; ct < ntiles; ct += 8) {
        const int j0 = ct << 4;

        // ---- phase 1: load interleaved pairs, sigmoid, stage tile in LDS ----
        {
            const float4* p =
                (const float4*)(bf + ((size_t)(i0 + r) * N + (size_t)(j0 + h * 8)) * 2);
            // prefetch 256 B ahead in this row (tile ct+2) into cache (global_prefetch_b8)
            if (ct + 8 < ntiles)
                __builtin_prefetch((const char*)(p) + 16 * 16, 0, 1);
            if (lane < 16) wb[wave][lane] = 1.0f - ap[j0 + lane];
#pragma unroll
            for (int q = 0; q < 4; ++q) {
                float4 t4 = p[q];              // two (ch0,ch1) pairs
                int c = h * 8 + q * 2;
                float cb0 = 1.0f - ash[j0 + c];
                float cb1 = 1.0f - ash[j0 + c + 1];
                tile[wave][r][c]     = sigmoidf((t4.y - t4.x) - sh_r * cb0);
                tile[wave][r][c + 1] = sigmoidf((t4.w - t4.z) - sh_r * cb1);
            }
        }
        __syncthreads();

        // ---- phase 2: dual reduction via V_WMMA_F32_16X16X4_F32 ----
        v8f cv = {};
#pragma unroll
        for (int c = 0; c < 4; ++c) {
            const int kb = 4 * c + koff;
            // u-path: A[m][k] = G[m][kb..kb+1] * b[j0+kb..]; B = ones(4x16)
            v2f A;
            A.x = tile[wave][m16][kb]     * wb[wave][kb];
            A.y = tile[wave][m16][kb + 1] * wb[wave][kb + 1];
            cu = __builtin_amdgcn_wmma_f32_16x16x4_f32(
                    false, A, false, ones, (short)0, cu, false, false);
            // v-path: B[k][n] = a[i0+kb..] * G[kb..][n]; A = ones(16x4)
            v2f B;
            B.x = tile[wave][kb][m16]     * arow[2 * c];
            B.y = tile[wave][kb + 1][m16] * arow[2 * c + 1];
            cv = __builtin_amdgcn_wmma_f32_16x16x4_f32(
                    false, ones, false, B, (short)0, cv, false, false);
        }
        // C/D layout: VGPR0 lanes0-15 = (M=0, N=lane); all rows identical here.
        if (lane < 16) atomicAdd(&v[j0 + lane], cv[0]);
        __syncthreads();
    }

    // extract u: VGPR rr holds rows rr (lanes 0-15) and rr+8 (lanes 16-31)
#pragma unroll
    for (int rr = 0; rr < 8; ++rr) {
        float val = cu[rr];
        if (lane == 0)  atomicAdd(&u[i0 + rr],     val);
        if (lane == 16) atomicAdd(&u[i0 + 8 + rr], val);
    }
}

// ---------------------------------------------------------------------------
// update: cur_s = base_s + w0*(u,v); a_t = sigmoid(cs1-cs0); reset u,v.
// At t==5 also emits cur_s into d_out.
// ---------------------------------------------------------------------------
__global__ void lmp_update(const float* __restrict__ bs,
                           const float* __restrict__ wp,
                           float* __restrict__ u, float* __restrict__ v,
                           float* __restrict__ anew,
                           float* __restrict__ outs /* nullable */, int N) {
    int i = blockIdx.x * blockDim.x + threadIdx.x;
    if (i < N) {
        float w0 = wp[0];
        float c0 = bs[2 * i]     + w0 * u[i];
        float c1 = bs[2 * i + 1] + w0 * v[i];
        anew[i] = sigmoidf(c1 - c0);
        if (outs != nullptr) { outs[2 * i] = c0; outs[2 * i + 1] = c1; }
        u[i] = 0.0f;
        v[i] = 0.0f;
    }
}

// ---------------------------------------------------------------------------
// final: out_f[i,j,0] = bf[i,j,0] + w0*a4[i]*(1-a4[j]); ch1 copied.
// Pure streaming: b128 load/modify/store, 4 (i,j) pairs per thread.
// ---------------------------------------------------------------------------
__global__ __launch_bounds__(256) void lmp_final_f(
        const float* __restrict__ bf, const float* __restrict__ a4,
        const float* __restrict__ wp, float* __restrict__ outf, int N) {
    size_t tid = (size_t)blockIdx.x * blockDim.x + threadIdx.x;
    size_t p = tid * 4;                         // first of 4 pairs (same row)
    if (p >= (size_t)N * N) return;
    float w0 = wp[0];
    size_t i = p / (size_t)N;
    size_t j = p % (size_t)N;
    float wa = w0 * a4[i];
    const float4* src = (const float4*)(bf + p * 2);
    float4* dst = (float4*)(outf + p * 2);
    float4 x0 = src[0], x1 = src[1];
    x0.x += wa * (1.0f - a4[j]);
    x0.z += wa * (1.0f - a4[j + 1]);
    x1.x += wa * (1.0f - a4[j + 2]);
    x1.z += wa * (1.0f - a4[j + 3]);
    dst[0] = x0;
    dst[1] = x1;
}

extern "C" void kernel_launch(void* const* d_in, const int* in_sizes, int n_in,
                              void* d_out, int out_size, void* d_ws, size_t ws_size,
                              hipStream_t stream) {
    const int N = in_sizes[0] / 2;                    // logits_smoke is (N,2)
    const float* bs = (const float*)d_in[0];
    const float* bf = (const float*)d_in[1];
    const float* wp = (const float*)d_in[2];
    float* out = (float*)d_out;                       // [cur_s (2N) | cur_f (2N^2)]
    float* ws  = (float*)d_ws;

    float* u   = ws;
    float* v   = ws + (size_t)N;
    float* zer = ws + 2 * (size_t)N;
    float* a[6];
    for (int t = 0; t < 6; ++t) a[t] = ws + (3 + (size_t)t) * N;

    const int tb = 256;
    lmp_init<<<(N + tb - 1) / tb, tb, 0, stream>>>(bs, a[0], u, v, zer, N);

    for (int t = 1; t <= 5; ++t) {
        const float* ash = (t >= 2) ? a[t - 2] : zer;
        lmp_matvec<<<N / 16, tb, 0, stream>>>(bf, a[t - 1], ash, wp, u, v, N);
        lmp_update<<<(N + tb - 1) / tb, tb, 0, stream>>>(
            bs, wp, u, v, a[t], (t == 5) ? out : nullptr, N);
    }

    size_t npairs  = (size_t)N * N;
    size_t nthread = npairs / 4;
    lmp_final_f<<<(unsigned)((nthread + tb - 1) / tb), tb, 0, stream>>>(
        bf, a[4], wp, out + 2 * (size_t)N, N);
}